// PrefixBLSTM_55001351192735
// MI455X (gfx1250) — compile-verified
//
#include <hip/hip_runtime.h>
#include <hip/hip_bf16.h>

// PrefixBLSTM on MI455X (gfx1250).
// fp32 WMMA (V_WMMA_F32_16X16X4_F32) + async global->LDS double-buffered GEMMs.
// Workspace requirement: ~4.1 MB (1,018,880 floats).

#define L_SEQ 128
#define T_SEQ 127
#define E_DIM 256
#define H_DIM 512
#define G_DIM 2048   // 4*H
#define V_DIM 32000

#define KC    16     // K-chunk staged per double-buffer step
#define LDSPAD 20    // row stride in floats (80B = 5*16B: aligned + conflict-free)

typedef float v2f __attribute__((ext_vector_type(2)));
typedef float v8f __attribute__((ext_vector_type(8)));
typedef int   v4i __attribute__((ext_vector_type(4)));

#define AS1 __attribute__((address_space(1)))
#define AS3 __attribute__((address_space(3)))

__device__ __forceinline__ float sigm(float x) { return 1.0f / (1.0f + __expf(-x)); }

// ---- CDNA5 async copy: 16B global -> LDS per lane (ASYNCcnt-tracked) -------
__device__ __forceinline__ void async_cp16(const float* gsrc, float* ldst) {
  // generic LDS pointer carries the LDS byte offset in its low 32 bits
  unsigned lds_off = (unsigned)(unsigned long long)ldst;
#if __has_builtin(__builtin_amdgcn_global_load_async_to_lds_b128)
  __builtin_amdgcn_global_load_async_to_lds_b128(
      (AS1 v4i*)(unsigned long long)gsrc, (AS3 v4i*)lds_off, 0, 0);
#else
  asm volatile("global_load_async_to_lds_b128 %0, %1, off"
               :: "v"(lds_off), "v"(gsrc) : "memory");
#endif
}

__device__ __forceinline__ void wait_async0() {
#if __has_builtin(__builtin_amdgcn_s_wait_asynccnt)
  __builtin_amdgcn_s_wait_asynccnt(0);
#else
  asm volatile("s_wait_asynccnt 0x0" ::: "memory");
#endif
}

// ---------------------------------------------------------------------------
// Init: gather embedding rows, zero LSTM states and padding rows.
// ---------------------------------------------------------------------------
__global__ void init_gather(const int* __restrict__ x, const float* __restrict__ emb_tab,
                            float* __restrict__ emb, float* __restrict__ Hb,
                            float* __restrict__ Cb, float* __restrict__ hf,
                            float* __restrict__ hstate, float* __restrict__ cstate) {
  int idx = blockIdx.x * 256 + threadIdx.x;
  if (idx < L_SEQ * E_DIM) {
    int t = idx / E_DIM, e = idx % E_DIM;
    emb[idx] = emb_tab[(long)x[t] * E_DIM + e];
    return;
  }
  idx -= L_SEQ * E_DIM;
  if (idx < 128 * H_DIM) { Hb[idx] = 0.0f; return; }
  idx -= 128 * H_DIM;
  if (idx < 128 * H_DIM) { Cb[idx] = 0.0f; return; }
  idx -= 128 * H_DIM;
  if (idx < H_DIM) { hstate[idx] = 0.0f; return; }
  idx -= H_DIM;
  if (idx < H_DIM) { cstate[idx] = 0.0f; return; }
  idx -= H_DIM;
  if (idx < H_DIM) { hf[(long)T_SEQ * H_DIM + idx] = 0.0f; return; }
}

// ---------------------------------------------------------------------------
// LDS-pipelined fp32 WMMA GEMM: C(M,N) = A(M,K) * Bt(N,K)^T  (+ optional bias)
//   FUSED_A: A(m,k) = k<512 ? A0[m*512+k] : A1[m*512+(k-512)]   ([hf|hb])
// Block = 256 threads = 8 waves; wave tile 16x64, block tile 32x256.
// Grid: (N/256, M/32). K % 16 == 0. Double-buffered async global->LDS staging.
// ---------------------------------------------------------------------------
template <int K, bool FUSED_A, bool BIAS>
__global__ __launch_bounds__(256) void wmma_gemm_lds(const float* __restrict__ A0,
                                                     const float* __restrict__ A1,
                                                     const float* __restrict__ Bt,
                                                     const float* __restrict__ bias,
                                                     float* __restrict__ C,
                                                     int N, int Mvalid) {
  __shared__ float Asmem[2][32 * LDSPAD];
  __shared__ float Bsmem[2][256 * LDSPAD];

  const int tid  = threadIdx.x;
  const int lane = tid & 31;
  const int wave = tid >> 5;
  const int wm = wave >> 2;           // 0..1
  const int wn = wave & 3;            // 0..3
  const int half = lane >> 4;         // 0..1
  const int l = lane & 15;            // 0..15
  const int rowBase = blockIdx.y * 32;
  const int colBase = blockIdx.x * 256;
  constexpr int NC = K / KC;

  // stage chunk c into buffer buf (async; each lane owns disjoint 16B pieces)
  auto stage = [&](int c, int buf) {
    const int k0 = c * KC;
    if (tid < 128) {                          // A tile: 32 rows x 16 floats
      int m = tid >> 2, seg = tid & 3;
      const float* gsrc;
      if (FUSED_A) {
        int mg = rowBase + m;
        gsrc = (k0 < H_DIM) ? A0 + (long)mg * H_DIM + k0 + seg * 4
                            : A1 + (long)mg * H_DIM + (k0 - H_DIM) + seg * 4;
      } else {
        gsrc = A0 + (long)(rowBase + m) * K + k0 + seg * 4;
      }
      async_cp16(gsrc, &Asmem[buf][m * LDSPAD + seg * 4]);
    }
#pragma unroll
    for (int i = 0; i < 4; ++i) {             // B tile: 256 rows x 16 floats
      int idx = tid + 256 * i;
      int n = idx >> 2, seg = idx & 3;
      const float* gsrc = Bt + (long)(colBase + n) * K + k0 + seg * 4;
      async_cp16(gsrc, &Bsmem[buf][n * LDSPAD + seg * 4]);
    }
  };

  v8f acc0 = {}, acc1 = {}, acc2 = {}, acc3 = {};
  stage(0, 0);

  for (int c = 0; c < NC; ++c) {
    wait_async0();          // this wave's chunk-c transfers landed
    __syncthreads();        // everyone's did
    if (c + 1 < NC) stage(c + 1, (c + 1) & 1);

    const float* As = &Asmem[c & 1][0];
    const float* Bs = &Bsmem[c & 1][0];
    const float* ap = As + (wm * 16 + l) * LDSPAD + 2 * half;
    const float* b0 = Bs + (wn * 64 + 0  + l) * LDSPAD + 2 * half;
    const float* b1 = Bs + (wn * 64 + 16 + l) * LDSPAD + 2 * half;
    const float* b2 = Bs + (wn * 64 + 32 + l) * LDSPAD + 2 * half;
    const float* b3 = Bs + (wn * 64 + 48 + l) * LDSPAD + 2 * half;

#pragma unroll
    for (int kk = 0; kk < KC; kk += 4) {
      v2f a  = *(const v2f*)(ap + kk);
      v2f vb0 = *(const v2f*)(b0 + kk);
      v2f vb1 = *(const v2f*)(b1 + kk);
      v2f vb2 = *(const v2f*)(b2 + kk);
      v2f vb3 = *(const v2f*)(b3 + kk);
      acc0 = __builtin_amdgcn_wmma_f32_16x16x4_f32(false, a, false, vb0, (short)0, acc0, false, false);
      acc1 = __builtin_amdgcn_wmma_f32_16x16x4_f32(false, a, false, vb1, (short)0, acc1, false, false);
      acc2 = __builtin_amdgcn_wmma_f32_16x16x4_f32(false, a, false, vb2, (short)0, acc2, false, false);
      acc3 = __builtin_amdgcn_wmma_f32_16x16x4_f32(false, a, false, vb3, (short)0, acc3, false, false);
    }
    __syncthreads();  // keep waves in lockstep before next overwrite window
  }

  // C/D layout: reg r -> M = rowBase + wm*16 + r + 8*half, N = colBase + wn*64 + j*16 + l
#pragma unroll
  for (int r = 0; r < 8; ++r) {
    int m = rowBase + wm * 16 + r + 8 * half;
    if (m < Mvalid) {
      int n = colBase + wn * 64 + l;
      float* Crow = C + (long)m * N + n;
      if (BIAS) {
        Crow[0]  = acc0[r] + bias[n];
        Crow[16] = acc1[r] + bias[n + 16];
        Crow[32] = acc2[r] + bias[n + 32];
        Crow[48] = acc3[r] + bias[n + 48];
      } else {
        Crow[0]  = acc0[r];
        Crow[16] = acc1[r];
        Crow[32] = acc2[r];
        Crow[48] = acc3[r];
      }
    }
  }
}

// ---------------------------------------------------------------------------
// Forward recurrent matvec: gf(2048) = W_hh_f(2048,512) @ h(512).
// ---------------------------------------------------------------------------
__global__ __launch_bounds__(256) void fwd_matvec(const float* __restrict__ Whh,
                                                  const float* __restrict__ h,
                                                  float* __restrict__ gf) {
  __shared__ float sh[H_DIM];
  for (int i = threadIdx.x; i < H_DIM; i += 256) sh[i] = h[i];
  __syncthreads();
  int g = blockIdx.x * 256 + threadIdx.x;
  const float* row = Whh + (long)g * H_DIM;
  float acc = 0.0f;
#pragma unroll 4
  for (int k = 0; k < H_DIM; k += 4) {
    float4 w = *(const float4*)(row + k);
    acc += w.x * sh[k] + w.y * sh[k + 1] + w.z * sh[k + 2] + w.w * sh[k + 3];
  }
  gf[g] = acc;
}

// ---------------------------------------------------------------------------
// Fused pointwise LSTM-cell update for step s.
//   blockIdx.y in [0,127): backward chain t (active iff s <= t)
//   blockIdx.y == 127    : forward cell update for t = s (writes hf[s])
// ---------------------------------------------------------------------------
__global__ __launch_bounds__(256) void lstm_update(int s,
    const float* __restrict__ G, const float* __restrict__ gx_b,
    const float* __restrict__ bib, const float* __restrict__ bhb,
    float* __restrict__ Hb, float* __restrict__ Cb,
    const float* __restrict__ gf, const float* __restrict__ gx_f,
    const float* __restrict__ bif, const float* __restrict__ bhf,
    float* __restrict__ hstate, float* __restrict__ cstate, float* __restrict__ hf) {
  const int u = blockIdx.x * 256 + threadIdx.x;  // 0..511
  const int t = blockIdx.y;                       // 0..127

  if (t == T_SEQ) {
    const float* gx = gx_f + (long)s * G_DIM;
    float gi = gx[u]        + gf[u]        + bif[u]        + bhf[u];
    float gF = gx[u + 512]  + gf[u + 512]  + bif[u + 512]  + bhf[u + 512];
    float gg = gx[u + 1024] + gf[u + 1024] + bif[u + 1024] + bhf[u + 1024];
    float go = gx[u + 1536] + gf[u + 1536] + bif[u + 1536] + bhf[u + 1536];
    float i = sigm(gi), f = sigm(gF), g = tanhf(gg), o = sigm(go);
    float c = f * cstate[u] + i * g;
    float h = o * tanhf(c);
    cstate[u] = c;
    hstate[u] = h;
    hf[(long)s * H_DIM + u] = h;
  } else if (s <= t) {
    const float* gx = gx_b + (long)(t - s) * G_DIM;
    const float* Gr = G + (long)t * G_DIM;
    float gi = Gr[u]        + gx[u]        + bib[u]        + bhb[u];
    float gF = Gr[u + 512]  + gx[u + 512]  + bib[u + 512]  + bhb[u + 512];
    float gg = Gr[u + 1024] + gx[u + 1024] + bib[u + 1024] + bhb[u + 1024];
    float go = Gr[u + 1536] + gx[u + 1536] + bib[u + 1536] + bhb[u + 1536];
    float i = sigm(gi), f = sigm(gF), g = tanhf(gg), o = sigm(go);
    long p = (long)t * H_DIM + u;
    float c = f * Cb[p] + i * g;
    float h = o * tanhf(c);
    Cb[p] = c;
    Hb[p] = h;
  }
}

// ---------------------------------------------------------------------------
extern "C" void kernel_launch(void* const* d_in, const int* in_sizes, int n_in,
                              void* d_out, int out_size, void* d_ws, size_t ws_size,
                              hipStream_t stream) {
  (void)in_sizes; (void)n_in; (void)out_size; (void)ws_size;
  const int*   x    = (const int*)d_in[0];
  const float* embt = (const float*)d_in[1];
  const float* Wif  = (const float*)d_in[2];
  const float* Whf  = (const float*)d_in[3];
  const float* bif  = (const float*)d_in[4];
  const float* bhf  = (const float*)d_in[5];
  const float* Wib  = (const float*)d_in[6];
  const float* Whb  = (const float*)d_in[7];
  const float* bib  = (const float*)d_in[8];
  const float* bhb  = (const float*)d_in[9];
  const float* fcw  = (const float*)d_in[10];
  const float* fcb  = (const float*)d_in[11];
  float* out = (float*)d_out;

  float* ws  = (float*)d_ws;
  float* emb = ws;                 // 128*256   = 32768
  float* gxf = emb + 32768;        // 128*2048  = 262144
  float* gxb = gxf + 262144;       // 128*2048  = 262144
  float* hf  = gxb + 262144;       // 128*512   = 65536  (row 127 = pad)
  float* Hb  = hf + 65536;         // 128*512   = 65536  (row 127 = pad)
  float* Cb  = Hb + 65536;         // 128*512   = 65536
  float* G   = Cb + 65536;         // 128*2048  = 262144
  float* gf  = G + 262144;         // 2048
  float* hst = gf + 2048;          // 512
  float* cst = hst + 512;          // 512  -> total 1,018,880 floats (~4.08 MB)

  // 1) gather + zero-init (exactly 165,376 work items)
  init_gather<<<646, 256, 0, stream>>>(x, embt, emb, Hb, Cb, hf, hst, cst);

  // 2) input-gate GEMMs: gx_f / gx_b = emb @ W_ih^T   (128 x 256 x 2048)
  wmma_gemm_lds<E_DIM, false, false><<<dim3(8, 4), 256, 0, stream>>>(
      emb, nullptr, Wif, nullptr, gxf, G_DIM, 128);
  wmma_gemm_lds<E_DIM, false, false><<<dim3(8, 4), 256, 0, stream>>>(
      emb, nullptr, Wib, nullptr, gxb, G_DIM, 128);

  // 3) 127 lockstep steps: backward batched recurrence (WMMA GEMM) +
  //    forward recurrence (matvec) + fused cell updates
  for (int s = 0; s < T_SEQ; ++s) {
    wmma_gemm_lds<H_DIM, false, false><<<dim3(8, 4), 256, 0, stream>>>(
        Hb, nullptr, Whb, nullptr, G, G_DIM, 128);
    fwd_matvec<<<8, 256, 0, stream>>>(Whf, hst, gf);
    lstm_update<<<dim3(2, 128), 256, 0, stream>>>(s, G, gxb, bib, bhb, Hb, Cb,
                                                  gf, gxf, bif, bhf, hst, cst, hf);
  }

  // 4) logits = [hf | Hb] @ fc_w^T + fc_b    (127 x 1024 x 32000)
  wmma_gemm_lds<2 * H_DIM, true, true><<<dim3(125, 4), 256, 0, stream>>>(
      hf, Hb, fcw, fcb, out, V_DIM, T_SEQ);
}